// model_activation_274877907762
// MI455X (gfx1250) — compile-verified
//
#include <hip/hip_runtime.h>
#include <math.h>

// ---------------------------------------------------------------------------
// MI455X (gfx1250) implementation.
// Bandwidth-bound problem (~640MB HBM traffic vs ~70 GFLOP) => f32 end-to-end,
// GEMMs on V_WMMA_F32_16X16X4_F32.  Each block owns FULL output rows so the
// per-row squared-norm reduces deterministically in-block (no FP atomics).
// A-tiles stream HBM->LDS via global_load_async_to_lds_b128 when available
// (mask folded into the epilogue: exact, since mask is a per-row 0/1 scalar).
// B-tiles are stored K-pair-interleaved in LDS so each WMMA B fragment is a
// single aligned ds_load_b64 (no VGPR repacking movs), with bank-skew pitches.
// Threshold selection = 4-pass radix select on float bit patterns (norms>=0).
// ---------------------------------------------------------------------------

typedef __attribute__((ext_vector_type(2))) float v2f;
typedef __attribute__((ext_vector_type(8))) float v8f;
typedef __attribute__((ext_vector_type(4))) int   v4i;

#define BATCH_N 65536
#define KTH_ASC 32767   // BATCH-1-round(BATCH*0.5): ascending rank of threshold

#if defined(__gfx1250__) && __has_builtin(__builtin_amdgcn_global_load_async_to_lds_b128) && __has_builtin(__builtin_amdgcn_s_wait_asynccnt)
#define USE_ASYNC_LDS 1
typedef __attribute__((address_space(1))) v4i* gv4i_p;   // global 16B vector
typedef __attribute__((address_space(3))) v4i* lv4i_p;   // LDS 16B vector
#else
#define USE_ASYNC_LDS 0
#endif

// ---------------------------------------------------------------------------
// Fused GEMM + prev-mask + bias + (ReLU) + per-row squared-L2-norm.
// Block = 256 threads = 8 waves; wave (wm,wn) computes rows [wm*16,wm*16+16)
// x cols [wn*WN, wn*WN+WN); block tile spans all N columns.
// N may be padded past NREAL (weights/bias zero-padded -> pad outputs are 0).
// ---------------------------------------------------------------------------
template<int K, int N, int NREAL, int WAVES_M, int WAVES_N, bool RELU>
__global__ __launch_bounds__(256)
void gemm_relu_norm(const float* __restrict__ A, int lda,
                    const float* __restrict__ W,
                    const float* __restrict__ bias,
                    const float* __restrict__ mask,   // may be nullptr
                    float* __restrict__ H, int ldh,
                    float* __restrict__ norms)
{
    constexpr int M_TILE = WAVES_M * 16;
    constexpr int WN     = N / WAVES_N;    // cols per wave
    constexpr int NT     = WN / 16;        // 16x16 wmma tiles per wave
    constexpr int APITCH = 20;             // 16 data + 4 skew (conflict-free)
    constexpr int PPITCH = 2 * N + 32;     // per-K-pair pitch, half-skewed

    __shared__ __align__(16) float lA[M_TILE * APITCH];
    __shared__ __align__(16) float lB[8 * PPITCH];
    __shared__ float lNorm[WAVES_N * M_TILE];

    const int tid  = threadIdx.x;
    const int wave = tid >> 5;
    const int lane = tid & 31;
    const int wm   = wave / WAVES_N;
    const int wn   = wave % WAVES_N;
    const int l16  = lane & 15;
    const int half = lane >> 4;            // 0: lanes 0-15, 1: lanes 16-31
    const int rowBase = blockIdx.x * M_TILE;

    v8f acc[NT] = {};

    for (int kc = 0; kc < K; kc += 16) {
        // --- stage A tile [M_TILE x 16] (pure copy; mask applied in epilogue)
#if USE_ASYNC_LDS
        for (int idx = tid; idx < M_TILE * 4; idx += 256) {
            int r  = idx >> 2;
            int c4 = idx & 3;
            const float* src = A + (size_t)(rowBase + r) * lda + kc + c4 * 4;
            float*       dst = lA + r * APITCH + c4 * 4;
            __builtin_amdgcn_global_load_async_to_lds_b128(
                (gv4i_p)src, (lv4i_p)dst, 0, 0);
        }
#else
        for (int idx = tid; idx < M_TILE * 4; idx += 256) {
            int r  = idx >> 2;
            int c4 = idx & 3;
            const float* src = A + (size_t)(rowBase + r) * lda + kc + c4 * 4;
            float4 v = *(const float4*)src;
            *(float4*)(lA + r * APITCH + c4 * 4) = v;
            if (kc + 16 < K) __builtin_prefetch(src + 16, 0, 1);
        }
#endif
        // --- stage B tile [16 x N], K-pair interleaved: pair p holds
        //     (B[2p][n], B[2p+1][n]) contiguously at lB[p*PPITCH + 2n] ---
        if constexpr (NREAL == N) {
            for (int idx = tid; idx < 8 * (N / 4); idx += 256) {
                int p  = idx / (N / 4);
                int n4 = idx % (N / 4);
                float4 r0 = *(const float4*)(W + (size_t)(kc + 2 * p)     * N + n4 * 4);
                float4 r1 = *(const float4*)(W + (size_t)(kc + 2 * p + 1) * N + n4 * 4);
                float* dst = lB + p * PPITCH + n4 * 8;
                ((float4*)dst)[0] = make_float4(r0.x, r1.x, r0.y, r1.y);
                ((float4*)dst)[1] = make_float4(r0.z, r1.z, r0.w, r1.w);
            }
        } else {  // zero-pad cols NREAL..N-1
            for (int idx = tid; idx < 8 * N; idx += 256) {
                int p = idx / N;
                int n = idx % N;
                v2f w;
                w.x = (n < NREAL) ? W[(size_t)(kc + 2 * p)     * NREAL + n] : 0.0f;
                w.y = (n < NREAL) ? W[(size_t)(kc + 2 * p + 1) * NREAL + n] : 0.0f;
                *(v2f*)(lB + p * PPITCH + n * 2) = w;
            }
        }
#if USE_ASYNC_LDS
        __builtin_amdgcn_s_wait_asynccnt(0);
#endif
        __syncthreads();

        // --- 4 wmma k-steps (K=4 each) over the 16-deep LDS tile ---
        for (int kk = 0; kk < 4; ++kk) {
            // A frag: lane l<16 holds K=kk*4+{0,1}; lanes 16-31 K=kk*4+{2,3}
            v2f a = *(const v2f*)(lA + (wm * 16 + l16) * APITCH + kk * 4 + half * 2);
            for (int nt = 0; nt < NT; ++nt) {
                // B frag: K-pair index = kk*2+half -> one aligned b64 load
                v2f b = *(const v2f*)(lB + (kk * 2 + half) * PPITCH
                                         + (wn * WN + nt * 16 + l16) * 2);
                acc[nt] = __builtin_amdgcn_wmma_f32_16x16x4_f32(
                    false, a, false, b, (short)0, acc[nt], false, false);
            }
        }
        __syncthreads();
    }

    // --- epilogue: prev-layer row mask (exact: m in {0,1}), bias, relu,
    //     store, per-row squared norm ---
    float rm[8];
    for (int v = 0; v < 8; ++v)
        rm[v] = mask ? mask[rowBase + wm * 16 + half * 8 + v] : 1.0f;

    float rowsum[8];
    for (int v = 0; v < 8; ++v) rowsum[v] = 0.0f;
    for (int nt = 0; nt < NT; ++nt) {
        int col = wn * WN + nt * 16 + l16;
        float bb = (col < NREAL) ? bias[col] : 0.0f;
        for (int v = 0; v < 8; ++v) {
            // C layout: VGPR v, lanes 0-15 -> M=v, lanes 16-31 -> M=8+v
            float c = acc[nt][v] * rm[v] + bb;
            if (RELU) c = fmaxf(c, 0.0f);
            rowsum[v] += c * c;
            int row = rowBase + wm * 16 + half * 8 + v;
            H[(size_t)row * ldh + col] = c;
        }
    }
    // reduce across the 16 lanes of each half (fixed order -> deterministic)
    for (int off = 8; off >= 1; off >>= 1)
        for (int v = 0; v < 8; ++v)
            rowsum[v] += __shfl_xor(rowsum[v], off, 16);
    if (l16 == 0)
        for (int v = 0; v < 8; ++v)
            lNorm[wn * M_TILE + wm * 16 + half * 8 + v] = rowsum[v];
    __syncthreads();
    // reduce across the WAVES_N column-strips (fixed order)
    for (int r = tid; r < M_TILE; r += 256) {
        float s = 0.0f;
        for (int q = 0; q < WAVES_N; ++q) s += lNorm[q * M_TILE + r];
        norms[rowBase + r] = s;     // squared L2 norm: order-equivalent
    }
}

// ---------------------------------------------------------------------------
// Single-block radix select: value of the kth-smallest (0-indexed ascending)
// of n non-negative floats, via 4x 8-bit digit passes on the bit pattern.
// ---------------------------------------------------------------------------
__global__ __launch_bounds__(1024)
void radix_select_kernel(const float* __restrict__ vals, int n, int kth,
                         float* __restrict__ Tout)
{
    __shared__ unsigned hist[256];
    __shared__ unsigned sh_prefix;
    __shared__ int      sh_krem;
    const int tid = threadIdx.x;
    if (tid == 0) { sh_prefix = 0u; sh_krem = kth; }
    __syncthreads();
    for (int p = 3; p >= 0; --p) {
        for (int i = tid; i < 256; i += 1024) hist[i] = 0u;
        __syncthreads();
        unsigned prefix = sh_prefix;
        unsigned hiMask = (p == 3) ? 0u : (0xFFFFFFFFu << (8 * (p + 1)));
        for (int i = tid; i < n; i += 1024) {
            unsigned u = __float_as_uint(vals[i]);
            if ((u & hiMask) == prefix)
                atomicAdd(&hist[(u >> (8 * p)) & 255u], 1u);
        }
        __syncthreads();
        if (tid == 0) {
            int krem = sh_krem;
            unsigned cum = 0;
            int d = 0;
            for (; d < 256; ++d) {
                unsigned c = hist[d];
                if (cum + c > (unsigned)krem) break;
                cum += c;
            }
            sh_prefix = prefix | ((unsigned)d << (8 * p));
            sh_krem   = krem - (int)cum;
        }
        __syncthreads();
    }
    if (tid == 0) Tout[0] = __uint_as_float(sh_prefix);
}

__global__ void mask_kernel(const float* __restrict__ norms,
                            const float* __restrict__ T,
                            float* __restrict__ mask, int n)
{
    int i = blockIdx.x * blockDim.x + threadIdx.x;
    if (i < n) mask[i] = (norms[i] > T[0]) ? 1.0f : 0.0f;
}

// Layer 4: h4 = (h3 * m3) @ W4 + b4  (10x10, no relu), plus squared norms.
__global__ __launch_bounds__(256)
void layer4_kernel(const float* __restrict__ h3,   // stride 16, 10 real cols
                   const float* __restrict__ m3,
                   const float* __restrict__ W4,    // [10,10]
                   const float* __restrict__ b4,    // [10]
                   float* __restrict__ h4,          // stride 16
                   float* __restrict__ norms)
{
    __shared__ float sW[100];
    __shared__ float sb[10];
    const int tid = threadIdx.x;
    if (tid < 100) sW[tid] = W4[tid];
    if (tid < 10)  sb[tid] = b4[tid];
    __syncthreads();
    int row = blockIdx.x * 256 + tid;
    float m = m3[row];
    float a[10];
    for (int i = 0; i < 10; ++i) a[i] = h3[(size_t)row * 16 + i] * m;
    float s = 0.0f;
    for (int j = 0; j < 10; ++j) {
        float c = sb[j];
        for (int i = 0; i < 10; ++i) c += a[i] * sW[i * 10 + j];
        h4[(size_t)row * 16 + j] = c;
        s += c * c;
    }
    norms[row] = s;
}

// Final: mask4 = norms4 > T4; out = softmax(h4 * mask4).
__global__ __launch_bounds__(256)
void softmax_kernel(const float* __restrict__ h4,     // stride 16
                    const float* __restrict__ norms,
                    const float* __restrict__ T,
                    float* __restrict__ mask4,
                    float* __restrict__ out)          // stride 10
{
    int row = blockIdx.x * 256 + threadIdx.x;
    float m = (norms[row] > T[0]) ? 1.0f : 0.0f;
    mask4[row] = m;
    float v[10];
    float mx = -INFINITY;
    for (int i = 0; i < 10; ++i) { v[i] = h4[(size_t)row * 16 + i] * m; mx = fmaxf(mx, v[i]); }
    float s = 0.0f;
    for (int i = 0; i < 10; ++i) { v[i] = expf(v[i] - mx); s += v[i]; }
    float inv = 1.0f / s;
    for (int i = 0; i < 10; ++i) out[(size_t)row * 10 + i] = v[i] * inv;
}

extern "C" void kernel_launch(void* const* d_in, const int* in_sizes, int n_in,
                              void* d_out, int out_size, void* d_ws, size_t ws_size,
                              hipStream_t stream)
{
    (void)in_sizes; (void)n_in; (void)out_size; (void)ws_size;
    const float* x  = (const float*)d_in[0];
    const float* W1 = (const float*)d_in[1];
    const float* b1 = (const float*)d_in[2];
    const float* W2 = (const float*)d_in[3];
    const float* b2 = (const float*)d_in[4];
    const float* W3 = (const float*)d_in[5];
    const float* b3 = (const float*)d_in[6];
    const float* W4 = (const float*)d_in[7];
    const float* b4 = (const float*)d_in[8];

    // d_out layout: out[65536*10] | m1 | m2 | m3 | m4 (each 65536 floats)
    float* out = (float*)d_out;
    float* m1  = out + (size_t)BATCH_N * 10;
    float* m2  = m1 + BATCH_N;
    float* m3  = m2 + BATCH_N;
    float* m4  = m3 + BATCH_N;

    // workspace carve-up (floats): ~210 MB
    float* ws    = (float*)d_ws;
    float* h1    = ws;                               // 65536 x 512
    float* h2    = h1 + (size_t)BATCH_N * 512;       // 65536 x 256
    float* h3    = h2 + (size_t)BATCH_N * 256;       // 65536 x 16 (10 real)
    float* h4    = h3 + (size_t)BATCH_N * 16;        // 65536 x 16 (10 real)
    float* norms = h4 + (size_t)BATCH_N * 16;        // 65536
    float* T     = norms + BATCH_N;                  // 1 float, reused

    // Layer 1: [65536,784]x[784,512]  (block tile 32 x 512)
    gemm_relu_norm<784, 512, 512, 2, 4, true>
        <<<BATCH_N / 32, 256, 0, stream>>>(x, 784, W1, b1, nullptr, h1, 512, norms);
    radix_select_kernel<<<1, 1024, 0, stream>>>(norms, BATCH_N, KTH_ASC, T);
    mask_kernel<<<BATCH_N / 256, 256, 0, stream>>>(norms, T, m1, BATCH_N);

    // Layer 2: [65536,512]x[512,256]  (block tile 64 x 256), mask m1 fused
    gemm_relu_norm<512, 256, 256, 4, 2, true>
        <<<BATCH_N / 64, 256, 0, stream>>>(h1, 512, W2, b2, m1, h2, 256, norms);
    radix_select_kernel<<<1, 1024, 0, stream>>>(norms, BATCH_N, KTH_ASC, T);
    mask_kernel<<<BATCH_N / 256, 256, 0, stream>>>(norms, T, m2, BATCH_N);

    // Layer 3: [65536,256]x[256,10->16]  (block tile 128 x 16), mask m2 fused
    gemm_relu_norm<256, 16, 10, 8, 1, true>
        <<<BATCH_N / 128, 256, 0, stream>>>(h2, 256, W3, b3, m2, h3, 16, norms);
    radix_select_kernel<<<1, 1024, 0, stream>>>(norms, BATCH_N, KTH_ASC, T);
    mask_kernel<<<BATCH_N / 256, 256, 0, stream>>>(norms, T, m3, BATCH_N);

    // Layer 4 (10x10) + final gate + softmax
    layer4_kernel<<<BATCH_N / 256, 256, 0, stream>>>(h3, m3, W4, b4, h4, norms);
    radix_select_kernel<<<1, 1024, 0, stream>>>(norms, BATCH_N, KTH_ASC, T);
    softmax_kernel<<<BATCH_N / 256, 256, 0, stream>>>(h4, norms, T, m4, out);
}